// Policy_38208029065712
// MI455X (gfx1250) — compile-verified
//
#include <hip/hip_runtime.h>
#include <hip/hip_bf16.h>

typedef __attribute__((ext_vector_type(2))) float v2f;
typedef __attribute__((ext_vector_type(8))) float v8f;

#define N_NODES 100000
#define N_EDGES 1600000
#define M_TILES (N_NODES / 16)   // 6250, exact

// ---------------- degree / normalization ----------------

__global__ __launch_bounds__(256) void deg_kernel(const int* __restrict__ dst,
                                                  float* __restrict__ deg, int E) {
    int e = blockIdx.x * 256 + threadIdx.x;
    if (e < E) atomicAdd(&deg[dst[e]], 1.0f);
}

__global__ __launch_bounds__(256) void dinv_kernel(float* __restrict__ deg, int n) {
    int i = blockIdx.x * 256 + threadIdx.x;
    if (i < n) deg[i] = rsqrtf(deg[i] + 1.0f);   // deg buffer becomes dinv_sqrt
}

// ---------------- f32 WMMA GEMM: out[N_NODES,NO] = A[N_NODES,K] @ W[K,NO] ----------------
// One wave computes one 16-row tile across all NO columns using V_WMMA_F32_16X16X4_F32.
// A frag (16x4): lanes 0-15 hold rows M=0..15 with K=k0,k0+1; lanes 16-31 same rows, K=k0+2,k0+3.
// B frag (4x16): VGPR0 = rows k0 (lanes 0-15) / k0+2 (lanes 16-31); VGPR1 = k0+1 / k0+3.
// D (16x16): VGPR j holds row j (lanes 0-15) and row j+8 (lanes 16-31).

template <int K, int NO>
__global__ __launch_bounds__(256) void gemm_wmma_kernel(const float* __restrict__ A,
                                                        const float* __restrict__ W,
                                                        float* __restrict__ out,
                                                        int m_tiles) {
    int wave = blockIdx.x * 8 + (threadIdx.x >> 5);
    if (wave >= m_tiles) return;            // wave-uniform: EXEC stays all-ones for WMMA
    int lane = threadIdx.x & 31;
    int lo = lane & 15;
    int hi = lane >> 4;
    int m_base = wave * 16;

    constexpr int NT = NO / 16;
    v8f acc[NT] = {};

    const float* arow = A + (long)(m_base + lo) * K;
#pragma unroll
    for (int k0 = 0; k0 < K; k0 += 4) {
        v2f a;
        a.x = arow[k0 + 2 * hi];
        a.y = arow[k0 + 2 * hi + 1];
#pragma unroll
        for (int t = 0; t < NT; ++t) {
            v2f b;
            b.x = W[(k0 + 2 * hi) * NO + t * 16 + lo];
            b.y = W[(k0 + 2 * hi + 1) * NO + t * 16 + lo];
            acc[t] = __builtin_amdgcn_wmma_f32_16x16x4_f32(
                false, a, false, b, (short)0, acc[t], false, false);
        }
    }

#pragma unroll
    for (int t = 0; t < NT; ++t)
#pragma unroll
        for (int j = 0; j < 8; ++j)
            out[(long)(m_base + j + 8 * hi) * NO + t * 16 + lo] = acc[t][j];
}

// ---------------- edge scatter: agg[dst] += h[src] * dinv[src]*dinv[dst] ----------------
// One thread per (edge, feature): 64/32 consecutive threads share an edge, so the
// h[src] gather and agg[dst] atomic are row-coalesced; node tables live in L2.

template <int F>
__global__ __launch_bounds__(256) void scatter_kernel(const int* __restrict__ src,
                                                      const int* __restrict__ dst,
                                                      const float* __restrict__ dinv,
                                                      const float* __restrict__ h,
                                                      float* __restrict__ agg,
                                                      int total) {
    int tid = blockIdx.x * 256 + threadIdx.x;
    if (tid >= total) return;
    int e = tid / F;
    int k = tid % F;
    int s = src[e];
    int d = dst[e];
    float nrm = dinv[s] * dinv[d];
    atomicAdd(&agg[(long)d * F + k], h[(long)s * F + k] * nrm);
}

// ---------------- finalize: agg = [relu](agg + h * dinv^2 + bias) ----------------

template <int F, bool RELU>
__global__ __launch_bounds__(256) void finalize_kernel(const float* __restrict__ h,
                                                       const float* __restrict__ dinv,
                                                       const float* __restrict__ bias,
                                                       float* __restrict__ agg,
                                                       int total) {
    int tid = blockIdx.x * 256 + threadIdx.x;
    if (tid >= total) return;
    int i = tid / F;
    int k = tid % F;
    float di = dinv[i];
    float v = agg[tid] + h[tid] * (di * di) + bias[k];
    agg[tid] = RELU ? fmaxf(v, 0.0f) : v;
}

// ---------------- per-edge dot: one wave32 per edge, shfl_xor reduction ----------------

__global__ __launch_bounds__(256) void edgedot_kernel(const int* __restrict__ src,
                                                      const int* __restrict__ dst,
                                                      const float* __restrict__ h,
                                                      float* __restrict__ out, int E) {
    int e = blockIdx.x * 8 + (threadIdx.x >> 5);
    if (e >= E) return;
    int lane = threadIdx.x & 31;
    int s = src[e];
    int d = dst[e];
    float v = h[(long)s * 32 + lane] * h[(long)d * 32 + lane];
#pragma unroll
    for (int off = 16; off > 0; off >>= 1) v += __shfl_xor(v, off, 32);
    if (lane == 0) out[e] = v;
}

// ---------------- launch ----------------

extern "C" void kernel_launch(void* const* d_in, const int* in_sizes, int n_in,
                              void* d_out, int out_size, void* d_ws, size_t ws_size,
                              hipStream_t stream) {
    const float* x  = (const float*)d_in[0];
    const int*   ei = (const int*)d_in[1];
    const float* W1 = (const float*)d_in[2];
    const float* b1 = (const float*)d_in[3];
    const float* W2 = (const float*)d_in[4];
    const float* b2 = (const float*)d_in[5];
    float* out = (float*)d_out;

    const int* src = ei;
    const int* dst = ei + N_EDGES;

    // workspace layout (floats)
    float* ws   = (float*)d_ws;
    float* dinv = ws;                       // N_NODES
    float* agg1 = dinv + N_NODES;           // N_NODES*64
    float* agg2 = agg1 + N_NODES * 64;      // N_NODES*32
    float* h1   = agg2 + N_NODES * 32;      // N_NODES*64
    float* h2   = h1 + N_NODES * 64;        // N_NODES*32

    // zero the atomic-accumulation targets (deg/dinv + agg1 + agg2) each call
    hipMemsetAsync(ws, 0, (size_t)(N_NODES + N_NODES * 64 + N_NODES * 32) * sizeof(float),
                   stream);

    // degrees -> dinv_sqrt
    deg_kernel<<<(N_EDGES + 255) / 256, 256, 0, stream>>>(dst, dinv, N_EDGES);
    dinv_kernel<<<(N_NODES + 255) / 256, 256, 0, stream>>>(dinv, N_NODES);

    // layer 1: h1 = x @ W1 (K=32, NO=64)
    gemm_wmma_kernel<32, 64><<<(M_TILES + 7) / 8, 256, 0, stream>>>(x, W1, h1, M_TILES);
    scatter_kernel<64><<<(N_EDGES * 64) / 256, 256, 0, stream>>>(src, dst, dinv, h1, agg1,
                                                                 N_EDGES * 64);
    finalize_kernel<64, true><<<(N_NODES * 64 + 255) / 256, 256, 0, stream>>>(
        h1, dinv, b1, agg1, N_NODES * 64);

    // layer 2: h2 = relu(h1') @ W2 (K=64, NO=32)
    gemm_wmma_kernel<64, 32><<<(M_TILES + 7) / 8, 256, 0, stream>>>(agg1, W2, h2, M_TILES);
    scatter_kernel<32><<<(N_EDGES * 32) / 256, 256, 0, stream>>>(src, dst, dinv, h2, agg2,
                                                                 N_EDGES * 32);
    finalize_kernel<32, false><<<(N_NODES * 32 + 255) / 256, 256, 0, stream>>>(
        h2, dinv, b2, agg2, N_NODES * 32);

    // per-edge logits
    edgedot_kernel<<<(N_EDGES + 7) / 8, 256, 0, stream>>>(src, dst, agg2, out, N_EDGES);
}